// PairwiseDist_71494025609703
// MI455X (gfx1250) — compile-verified
//
#include <hip/hip_runtime.h>

// SSN pairwise distance, MI455X (gfx1250), wave32.
// d[b,k,n] = ||pFea[b,:,n] - spFea[b,:,idx_k]||^2  via  ||p||^2 - 2 p.s + ||s||^2
// Memory-bound (~252 MB HBM traffic -> ~11 us roofline at 23.3 TB/s).
// spFea staged in LDS via Tensor Data Mover; ||s||^2 via fp32 WMMA Gram
// diagonals; pFea/out streamed coalesced with non-temporal hints.

typedef __attribute__((ext_vector_type(2))) float    v2f;
typedef __attribute__((ext_vector_type(8))) float    v8f;
typedef __attribute__((ext_vector_type(4))) unsigned v4u;
typedef __attribute__((ext_vector_type(8))) int      v8i;
typedef __attribute__((ext_vector_type(4))) int      v4i;

#define B_    8
#define C_    20
#define N_    262144
#define NSP   1024
#define NSPW  32
#define BLOCK 1024     // 32 wave32 waves -> fills one WGP (8 waves/SIMD32)
#define PPT   8        // pixels per thread

#define RAW_F (C_ * NSP)   // 20480 floats = 80 KB   spFea[b], layout [c][j]
#define AUG_F (NSP * 24)   // 24576 floats = 96 KB   aug[j][24] = {-2*s_c..., snorm, pad}
#define SN_F  (NSP)        //  1024 floats =  4 KB   ||s_j||^2

__global__ __launch_bounds__(BLOCK)
void ssn_pairwise_kernel(const float* __restrict__ pFea,
                         const float* __restrict__ spFea,
                         const int*   __restrict__ spIdx,
                         float* __restrict__ out) {
  __shared__ __align__(16) float smem[RAW_F + AUG_F + SN_F];  // 180 KB of 320 KB WGP LDS
  float* raw  = smem;                   // LDS byte offset 0 (single static alloc)
  float* aug  = smem + RAW_F;
  float* snrm = smem + RAW_F + AUG_F;

  const int tid  = threadIdx.x;
  const int lane = tid & 31;
  const int wave = tid >> 5;
  const int b    = blockIdx.y;

  const float* spB = spFea + (size_t)b * (C_ * NSP);

  // ---- Phase 1: Tensor Data Mover: async DMA spFea[b] (80 KB) global -> LDS ----
  if (wave == 0) {
    unsigned long long ga = (unsigned long long)(const void*)spB;
    v4u g0;
    g0[0] = 1u;                                   // count = 1 valid descriptor
    g0[1] = 0u;                                   // lds_addr = 0 (raw[] at LDS base)
    g0[2] = (unsigned)(ga & 0xFFFFFFFFull);       // global_addr[31:0]
    g0[3] = (unsigned)((ga >> 32) & 0x1FFFFFFull) // global_addr[56:32]
          | (2u << 30);                           // type = 2 ("image")
    v8i g1;
    g1[0] = (int)(2u << 16);                      // data_size = 2 -> 4-byte elements
    g1[1] = (int)((unsigned)RAW_F << 16);         // tensor_dim0[15:0] = 20480
    g1[2] = (int)(1u << 16);                      // tensor_dim0 hi = 0 | tensor_dim1 = 1
    g1[3] = (int)((unsigned)RAW_F << 16);         // tile_dim0 = 20480 (one full row)
    g1[4] = 1;                                    // tile_dim1 = 1, tile_dim2 = 0
    g1[5] = RAW_F;                                // tensor_dim0_stride[31:0]
    g1[6] = 0;                                    // stride0 hi / stride1 lo
    g1[7] = 0;
    v4i z4 = {0, 0, 0, 0};
    v8i z8 = {0, 0, 0, 0, 0, 0, 0, 0};
    __builtin_amdgcn_tensor_load_to_lds(g0, g1, z4, z4, z8, 0);  // 6-arg form (clang-23)
    __builtin_amdgcn_s_wait_tensorcnt(0);         // s_wait_tensorcnt 0
  }
  __syncthreads();

  // ---- Phase 2: ||s_j||^2 via V_WMMA_F32_16X16X4_F32 Gram diagonals ----
  // Wave computes D = S_tile * S_tile^T (16 superpixels x 20 channels, 5 K=4 steps).
  // A-layout (16x4 f32): lane L holds row M=L%16, K = 2*(L/16)+{0,1}.
  // B-layout mirrors A (B cols across lanes), and B = A^T here -> same registers.
  #pragma unroll
  for (int tt = 0; tt < 2; ++tt) {
    const int j0  = (wave * 2 + tt) * 16;        // 32 waves x 2 tiles = 64 tiles = 1024 sp
    const int m   = lane & 15;
    const int ksh = (lane >> 4) << 1;
    v8f acc = {};
    #pragma unroll
    for (int kk = 0; kk < 5; ++kk) {
      const int c0 = kk * 4 + ksh;
      v2f a;
      a.x = raw[(c0 + 0) * NSP + j0 + m];
      a.y = raw[(c0 + 1) * NSP + j0 + m];
      acc = __builtin_amdgcn_wmma_f32_16x16x4_f32(false, a, false, a,
                                                  (short)0, acc, false, false);
    }
    // C/D layout: VGPR r -> (M=r, N=lane) for lanes 0-15, (M=r+8, N=lane-16) for 16-31.
    // Diagonal element (r, r) lives at lane r; (r+8, r+8) at lane r+24.
    float diag = 0.0f;
    #pragma unroll
    for (int r = 0; r < 8; ++r) {
      float lo = __shfl(acc[r], r, 32);
      float hi = __shfl(acc[r], r + 24, 32);
      diag = (lane == r)     ? lo : diag;
      diag = (lane == r + 8) ? hi : diag;
    }
    if (lane < 16) snrm[j0 + lane] = diag;
  }
  __syncthreads();

  // ---- Phase 3: build augmented table aug[j] = {-2*s_0..-2*s_19, snorm_j, 0,0,0} ----
  {
    const int j = tid;                            // BLOCK == NSP == 1024
    float* aj = aug + j * 24;                     // 96 B stride, 16 B aligned
    #pragma unroll
    for (int c = 0; c < C_; ++c) aj[c] = -2.0f * raw[c * NSP + j];
    aj[20] = snrm[j];
    aj[21] = 0.0f; aj[22] = 0.0f; aj[23] = 0.0f;
  }
  __syncthreads();

  // ---- Phase 4: streamed pixel loop: d = pnorm + snorm_j + sum_c p_c * (-2 s_c) ----
  const float* pB = pFea  + (size_t)b * C_ * N_;
  const int*   iB = spIdx + (size_t)b * N_;
  float*       oB = out   + (size_t)b * 9 * N_;
  const int base  = blockIdx.x * (BLOCK * PPT);

  for (int it = 0; it < PPT; ++it) {
    const int n = base + it * BLOCK + tid;        // coalesced across the block

    float p[C_];
    #pragma unroll
    for (int c = 0; c < C_; ++c)
      p[c] = __builtin_nontemporal_load(pB + (size_t)c * N_ + n);

    float pn = 0.0f;
    #pragma unroll
    for (int c = 0; c < C_; ++c) pn = fmaf(p[c], p[c], pn);

    const int idx = iB[n];
    const int x = idx & (NSPW - 1);
    const int y = idx >> 5;

    #pragma unroll
    for (int k = 0; k < 9; ++k) {
      const int dx = k % 3 - 1, dy = k / 3 - 1;
      const int nx = x + dx,    ny = y + dy;
      const bool valid = ((unsigned)nx < (unsigned)NSPW) &&
                         ((unsigned)ny < (unsigned)NSPW);
      int j = ny * NSPW + nx;
      j = j < 0 ? 0 : (j > NSP - 1 ? NSP - 1 : j);  // matches reference clip

      const float4* av = (const float4*)(aug + j * 24);  // 5x ds_load_b128
      float4 v0 = av[0], v1 = av[1], v2 = av[2], v3 = av[3], v4 = av[4];
      float  sn = aug[j * 24 + 20];

      float acc = pn + sn;
      acc = fmaf(p[0],  v0.x, acc); acc = fmaf(p[1],  v0.y, acc);
      acc = fmaf(p[2],  v0.z, acc); acc = fmaf(p[3],  v0.w, acc);
      acc = fmaf(p[4],  v1.x, acc); acc = fmaf(p[5],  v1.y, acc);
      acc = fmaf(p[6],  v1.z, acc); acc = fmaf(p[7],  v1.w, acc);
      acc = fmaf(p[8],  v2.x, acc); acc = fmaf(p[9],  v2.y, acc);
      acc = fmaf(p[10], v2.z, acc); acc = fmaf(p[11], v2.w, acc);
      acc = fmaf(p[12], v3.x, acc); acc = fmaf(p[13], v3.y, acc);
      acc = fmaf(p[14], v3.z, acc); acc = fmaf(p[15], v3.w, acc);
      acc = fmaf(p[16], v4.x, acc); acc = fmaf(p[17], v4.y, acc);
      acc = fmaf(p[18], v4.z, acc); acc = fmaf(p[19], v4.w, acc);

      __builtin_nontemporal_store(valid ? acc : 1e16f,
                                  oB + (size_t)k * N_ + n);
    }
  }
}

extern "C" void kernel_launch(void* const* d_in, const int* in_sizes, int n_in,
                              void* d_out, int out_size, void* d_ws, size_t ws_size,
                              hipStream_t stream) {
  (void)in_sizes; (void)n_in; (void)d_ws; (void)ws_size; (void)out_size;
  const float* pFea  = (const float*)d_in[0];
  const float* spFea = (const float*)d_in[1];
  const int*   spIdx = (const int*)d_in[2];
  float*       out   = (float*)d_out;

  dim3 grid(N_ / (BLOCK * PPT), B_, 1);   // 32 x 8 blocks, 1024 threads each
  hipLaunchKernelGGL(ssn_pairwise_kernel, grid, dim3(BLOCK), 0, stream,
                     pFea, spFea, spIdx, out);
}